// Kwta_40673340293744
// MI455X (gfx1250) — compile-verified
//
#include <hip/hip_runtime.h>
#include <stdint.h>

#define KWTA_D 4096
#define KWTA_THREADS 256

// Map float bits to an order-preserving unsigned key:
// larger key <=> larger float (NaN-free input per reference: N(0,1)).
__device__ __forceinline__ unsigned kwta_key(float f) {
  unsigned u = __float_as_uint(f);
  unsigned m = (unsigned)((int)u >> 31) | 0x80000000u;
  return u ^ m;
}

__global__ __launch_bounds__(KWTA_THREADS) void kwta_kernel(
    const float* __restrict__ x, const int* __restrict__ kptr,
    float* __restrict__ out) {
  __shared__ __align__(16) float rowf[KWTA_D];   // 16 KB row staging
  __shared__ unsigned hist[KWTA_THREADS];        // 256-bin histogram / scan
  __shared__ unsigned sel_prefix;                // accumulated key prefix
  __shared__ unsigned sel_k;                     // remaining rank

  const unsigned tid = threadIdx.x;
  const unsigned row = blockIdx.x;
  const float* rowp = x + (size_t)row * KWTA_D;

  // ---- CDNA5 async copy: global -> LDS, tracked by ASYNCcnt ----
  // 4 iters x 256 lanes x 16B = 16 KB row. SADDR = uniform 64-bit row base,
  // VADDR = per-lane byte offset, VDST = per-lane LDS byte address.
  const unsigned long long sbase = (unsigned long long)(uintptr_t)rowp;
#pragma unroll
  for (int it = 0; it < 4; ++it) {
    unsigned fidx = (unsigned)it * (KWTA_THREADS * 4) + tid * 4;
    unsigned goff = fidx * 4u;  // byte offset within the row
    unsigned lds_addr = (unsigned)(uintptr_t)(&rowf[fidx]);  // low 32 = LDS offset
    asm volatile("global_load_async_to_lds_b128 %0, %1, %2"
                 :
                 : "v"(lds_addr), "v"(goff), "s"(sbase)
                 : "memory");
  }
  asm volatile("s_wait_asynccnt 0" ::: "memory");
  __syncthreads();

  if (tid == 0) {
    sel_prefix = 0u;
    sel_k = (unsigned)(*kptr);  // k lives in device memory (scalar input)
  }
  __syncthreads();

  // ---- Exact MSD radix select: 4 passes of 8-bit digits over 4096 keys ----
#pragma unroll 1
  for (int shift = 24; shift >= 0; shift -= 8) {
    hist[tid] = 0u;
    __syncthreads();

    const unsigned prefix  = sel_prefix;
    const unsigned want    = sel_k;
    const unsigned mask_hi = (shift == 24) ? 0u : (0xFFFFFFFFu << (shift + 8));

#pragma unroll
    for (int it = 0; it < 4; ++it) {
      int i = it * (KWTA_THREADS * 4) + (int)tid * 4;
      float4 f4 = *(const float4*)&rowf[i];  // ds_load_b128
      unsigned kk0 = kwta_key(f4.x), kk1 = kwta_key(f4.y);
      unsigned kk2 = kwta_key(f4.z), kk3 = kwta_key(f4.w);
      if ((kk0 & mask_hi) == prefix) atomicAdd(&hist[(kk0 >> shift) & 0xFFu], 1u);
      if ((kk1 & mask_hi) == prefix) atomicAdd(&hist[(kk1 >> shift) & 0xFFu], 1u);
      if ((kk2 & mask_hi) == prefix) atomicAdd(&hist[(kk2 >> shift) & 0xFFu], 1u);
      if ((kk3 & mask_hi) == prefix) atomicAdd(&hist[(kk3 >> shift) & 0xFFu], 1u);
    }
    __syncthreads();

    // Inclusive suffix scan over 256 bins: hist[b] = sum_{j>=b} count[j]
#pragma unroll
    for (int off = 1; off < KWTA_THREADS; off <<= 1) {
      unsigned t = (tid + off < KWTA_THREADS) ? hist[tid + off] : 0u;
      __syncthreads();
      hist[tid] += t;
      __syncthreads();
    }

    // Unique winner: largest digit b with suffix[b] >= want
    unsigned suf     = hist[tid];
    unsigned sufnext = (tid < KWTA_THREADS - 1) ? hist[tid + 1] : 0u;
    if (suf >= want && sufnext < want) {
      sel_prefix = prefix | (tid << shift);
      sel_k = want - sufnext;
    }
    __syncthreads();
  }

  const unsigned thresh = sel_prefix;  // exact key of the k-th largest element

  // ---- Threshold + coalesced b128 store back to global ----
  float* outp = out + (size_t)row * KWTA_D;
#pragma unroll
  for (int it = 0; it < 4; ++it) {
    int i = it * (KWTA_THREADS * 4) + (int)tid * 4;
    float4 f4 = *(const float4*)&rowf[i];
    float4 r;
    r.x = (kwta_key(f4.x) >= thresh) ? f4.x : 0.0f;
    r.y = (kwta_key(f4.y) >= thresh) ? f4.y : 0.0f;
    r.z = (kwta_key(f4.z) >= thresh) ? f4.z : 0.0f;
    r.w = (kwta_key(f4.w) >= thresh) ? f4.w : 0.0f;
    *(float4*)&outp[i] = r;
  }
}

extern "C" void kernel_launch(void* const* d_in, const int* in_sizes, int n_in,
                              void* d_out, int out_size, void* d_ws, size_t ws_size,
                              hipStream_t stream) {
  const float* x = (const float*)d_in[0];
  const int* kptr = (const int*)d_in[1];
  float* out = (float*)d_out;
  int rows = in_sizes[0] / KWTA_D;  // 8192
  kwta_kernel<<<dim3(rows), dim3(KWTA_THREADS), 0, stream>>>(x, kptr, out);
}